// QuantumGenerator_24919400251795
// MI455X (gfx1250) — compile-verified
//
#include <hip/hip_runtime.h>
#include <hip/hip_bf16.h>

#define N_QUBITS 5
#define N_LAYERS 3
#define DIM 32           // 2^5 amplitudes
#define OUT_DIM 10

typedef float v2f __attribute__((ext_vector_type(2)));
typedef float v8f __attribute__((ext_vector_type(8)));

// ---------------------------------------------------------------------------
// Kernel 1: build the 32x32 circuit unitary from params.
// One wave (32 threads); thread j applies the full gate sequence to basis
// vector e_j, producing column j of U. Stored to workspace as W[64][32]:
// rows 0..31 = Re(U), rows 32..63 = Im(U), row-major (row = out amp, col = k).
// ---------------------------------------------------------------------------
__global__ __launch_bounds__(32) void build_unitary_kernel(
    const float* __restrict__ params, float* __restrict__ W) {
  const int col = threadIdx.x;  // basis column 0..31
  float sr[DIM], si[DIM];
#pragma unroll
  for (int i = 0; i < DIM; ++i) { sr[i] = (i == col) ? 1.0f : 0.0f; si[i] = 0.0f; }

  for (int layer = 0; layer < N_LAYERS; ++layer) {
#pragma unroll
    for (int q = 0; q < N_QUBITS; ++q) {
      const int mask = 1 << (4 - q);              // qubit q bit position (MSB-first)
      const float* pg = params + (layer * N_QUBITS + q) * 3;
      // ---- RX(theta): [[c, -i s], [-i s, c]] ----
      {
        float h = pg[0] * 0.5f, c = __cosf(h), s = __sinf(h);
#pragma unroll
        for (int i = 0; i < DIM; ++i) {
          if (i & mask) continue;
          int i1 = i | mask;
          float a0r = sr[i], a0i = si[i], a1r = sr[i1], a1i = si[i1];
          sr[i]  = c * a0r + s * a1i;
          si[i]  = c * a0i - s * a1r;
          sr[i1] = c * a1r + s * a0i;
          si[i1] = c * a1i - s * a0r;
        }
      }
      // ---- RY(theta): [[c, -s], [s, c]] ----
      {
        float h = pg[1] * 0.5f, c = __cosf(h), s = __sinf(h);
#pragma unroll
        for (int i = 0; i < DIM; ++i) {
          if (i & mask) continue;
          int i1 = i | mask;
          float a0r = sr[i], a0i = si[i], a1r = sr[i1], a1i = si[i1];
          sr[i]  = c * a0r - s * a1r;
          si[i]  = c * a0i - s * a1i;
          sr[i1] = s * a0r + c * a1r;
          si[i1] = s * a0i + c * a1i;
        }
      }
      // ---- RZ(theta): diag(e^{-ih}, e^{+ih}) ----
      {
        float h = pg[2] * 0.5f, c = __cosf(h), s = __sinf(h);
#pragma unroll
        for (int i = 0; i < DIM; ++i) {
          if (i & mask) continue;
          int i1 = i | mask;
          float a0r = sr[i], a0i = si[i], a1r = sr[i1], a1i = si[i1];
          sr[i]  = c * a0r + s * a0i;
          si[i]  = c * a0i - s * a0r;
          sr[i1] = c * a1r - s * a1i;
          si[i1] = c * a1i + s * a1r;
        }
      }
    }
    // ---- CNOT chain: control q, target q+1 ----
#pragma unroll
    for (int q = 0; q < N_QUBITS - 1; ++q) {
      const int mc = 1 << (4 - q);
      const int mt = 1 << (3 - q);
#pragma unroll
      for (int i = 0; i < DIM; ++i) {
        if (!(i & mc) || (i & mt)) continue;     // control=1, target=0
        int i1 = i | mt;
        float tr = sr[i]; sr[i] = sr[i1]; sr[i1] = tr;
        float ti = si[i]; si[i] = si[i1]; si[i1] = ti;
      }
    }
  }
#pragma unroll
  for (int i = 0; i < DIM; ++i) {
    W[i * DIM + col]         = sr[i];   // Re rows 0..31
    W[(DIM + i) * DIM + col] = si[i];   // Im rows 32..63
  }
}

// ---------------------------------------------------------------------------
// Kernel 2: batched state-prep + WMMA GEMM Out[64xB] = W[64x32] * S[32xB],
// then probability / signed reduction to z_q, writing out[B][10].
// One wave handles a 16-element batch tile; 4 m-tiles x 8 k-chunks = 32
// v_wmma_f32_16x16x4_f32 per tile; grid-stride over tiles (wave-uniform).
// ---------------------------------------------------------------------------
__global__ __launch_bounds__(256) void qgen_wmma_kernel(
    const float* __restrict__ x, const float* __restrict__ W,
    float* __restrict__ out, int nTiles) {
  const int lane = threadIdx.x & 31;
  const int lo = lane & 15;       // M row / batch column within tile
  const int hi = lane >> 4;       // selects K pair within a k-chunk
  const int waveGlobal = blockIdx.x * (blockDim.x >> 5) + (threadIdx.x >> 5);
  const int totalWaves = gridDim.x * (blockDim.x >> 5);

  // ---- Load A fragments once per wave (ISA 16x4 f32 A layout):
  // lane<16: (M=lo, K=4kc+{0,1}); lane>=16: (M=lo, K=4kc+{2,3})
  v2f a[4][8];
#pragma unroll
  for (int mt = 0; mt < 4; ++mt)
#pragma unroll
    for (int kc = 0; kc < 8; ++kc)
      a[mt][kc] = *(const v2f*)(W + (mt * 16 + lo) * DIM + kc * 4 + 2 * hi);

  for (int tile = waveGlobal; tile < nTiles; tile += totalWaves) {
    const int n = tile * 16 + lo;   // this lane's batch element
    // ---- angles: v[q] = (cos, sin)(x_q * pi/2); lanes l and l+16 duplicate
    float c[N_QUBITS], s[N_QUBITS];
#pragma unroll
    for (int q = 0; q < N_QUBITS; ++q) {
      float ang = x[n * N_QUBITS + q] * 1.57079632679489662f;
      c[q] = __cosf(ang);
      s[q] = __sinf(ang);
    }
    // ---- B fragments (4x16 f32 B layout, mirrors A): lane holds
    // S[k][lo] for k = 4kc + 2*hi + j ; s_k = prod_q (bit_q(k) ? sin : cos)
    v2f b[8];
#pragma unroll
    for (int kc = 0; kc < 8; ++kc) {
#pragma unroll
      for (int j = 0; j < 2; ++j) {
        const int k = kc * 4 + 2 * hi + j;
        float p = 1.0f;
#pragma unroll
        for (int q = 0; q < N_QUBITS; ++q)
          p *= ((k >> (4 - q)) & 1) ? s[q] : c[q];
        b[kc][j] = p;
      }
    }
    // ---- GEMM: acc[0..1] = Re rows 0..31, acc[2..3] = Im rows 0..31
    v8f acc[4] = {};
#pragma unroll
    for (int kc = 0; kc < 8; ++kc)
#pragma unroll
      for (int mt = 0; mt < 4; ++mt)
        acc[mt] = __builtin_amdgcn_wmma_f32_16x16x4_f32(
            false, a[mt][kc], false, b[kc], (short)0, acc[mt], false, false);

    // ---- probs + signed reduction. C/D layout: VGPR r holds row r (lanes
    // 0-15) and row r+8 (lanes 16-31), column = lo. Lane owns amps
    // iA = 8*hi + r (acc0/acc2) and iB = iA + 16 (acc1/acc3).
    float pz[N_QUBITS] = {0.f, 0.f, 0.f, 0.f, 0.f};
#pragma unroll
    for (int r = 0; r < 8; ++r) {
      const int iA = hi * 8 + r;
      const int iB = iA + 16;
      float pA = acc[0][r] * acc[0][r] + acc[2][r] * acc[2][r];
      float pB = acc[1][r] * acc[1][r] + acc[3][r] * acc[3][r];
#pragma unroll
      for (int q = 0; q < N_QUBITS; ++q) {
        pz[q] += ((iA >> (4 - q)) & 1) ? -pA : pA;
        pz[q] += ((iB >> (4 - q)) & 1) ? -pB : pB;
      }
    }
    // ---- combine the two half-rows (lane n <-> lane n+16)
#pragma unroll
    for (int q = 0; q < N_QUBITS; ++q)
      pz[q] += __shfl_xor(pz[q], 16, 32);

    if (lane < 16) {
      float* o = out + (size_t)n * OUT_DIM;
#pragma unroll
      for (int q = 0; q < N_QUBITS; ++q) o[q] = pz[q];
#pragma unroll
      for (int q = N_QUBITS; q < OUT_DIM; ++q) o[q] = 0.0f;
    }
  }
}

extern "C" void kernel_launch(void* const* d_in, const int* in_sizes, int n_in,
                              void* d_out, int out_size, void* d_ws, size_t ws_size,
                              hipStream_t stream) {
  const float* x = (const float*)d_in[0];       // [B][5]
  const float* params = (const float*)d_in[1];  // [45]
  float* out = (float*)d_out;                   // [B][10]
  float* W = (float*)d_ws;                      // [64][32] = 8 KB scratch

  const int batch = in_sizes[0] / N_QUBITS;     // 131072
  const int nTiles = batch / 16;                // 8192 (batch is 16-aligned)

  build_unitary_kernel<<<1, 32, 0, stream>>>(params, W);

  int blocks = 256;                             // 2048 waves, ~4 tiles/wave
  if (blocks * 8 > nTiles) blocks = (nTiles + 7) / 8;
  qgen_wmma_kernel<<<blocks, 256, 0, stream>>>(x, W, out, nTiles);
}